// AdvancedRegressionModel_65481071397239
// MI455X (gfx1250) — compile-verified
//
#include <hip/hip_runtime.h>

// ============================================================================
// Quantum circuit regression for MI455X (gfx1250), compile-only tuned.
//
// out[b] = b0 + sum_i W[i]*feats[i],  feats[i] = <psi| Zprefix_i |psi>,
// psi = (U_0 (x) ... (x) U_19) state, U_i = RY(p[i+20])RX(p[i]).
// CNOT ladder folded into the reduction signs (it is a basis permutation).
//
// Two HBM passes using 320KB-WGP LDS closures (vs 5 register-only passes):
//   pass1: wires 0..7  (two 4-wire WMMA stages through 32KB LDS)
//          real input (128MB read) -> complex state (256MB write)
//   pass2: wires 8..19 (three 4-wire WMMA stages in 32KB LDS) + fused
//          signed |psi|^2 reduction (256MB read, no state write)
// Total ~640MB -> ~27us at 23.3 TB/s.   Needs ws >= 1MB + 256MB.
// ============================================================================

typedef __attribute__((ext_vector_type(2))) float v2f;
typedef __attribute__((ext_vector_type(8))) float v8f;

#define NW     20
#define QBATCH 32

__device__ __forceinline__ v8f wmma_f32(v2f a, v2f b, v8f c) {
  // D(16x16,f32) += A(16x4,f32) * B(4x16,f32)
  return __builtin_amdgcn_wmma_f32_16x16x4_f32(
      false, a, false, b, (short)0, c, false, false);
}

// LDS skew: kills the 16-way bank conflict of the stride-16 stage-3 reads.
__device__ __forceinline__ unsigned SK(unsigned i) { return i + (i >> 4); }

struct AFrag { v2f r[4], i[4], ni[4]; };

// A-fragment (16x4 f32 layout: lanes = M row, VGPR x lane-half = K column)
__device__ __forceinline__ void loadA(const float* __restrict__ Gr,
                                      const float* __restrict__ Gi,
                                      int t, int hk, AFrag& A) {
  #pragma unroll
  for (int k = 0; k < 4; ++k) {
    int s0 = 4 * k + hk;
    float r0 = Gr[t * 16 + s0], r1 = Gr[t * 16 + s0 + 1];
    float i0 = Gi[t * 16 + s0], i1 = Gi[t * 16 + s0 + 1];
    A.r[k].x  =  r0; A.r[k].y  =  r1;
    A.i[k].x  =  i0; A.i[k].y  =  i1;
    A.ni[k].x = -i0; A.ni[k].y = -i1;  // f32 WMMA has no A-negate modifier
  }
}

// ---------------------------------------------------------------------------
// Five 16x16 complex group matrices G_p = U_{4p} (x) ... (x) U_{4p+3}
// ---------------------------------------------------------------------------
__global__ void setup_gates(const float* __restrict__ params,
                            float* __restrict__ Gr, float* __restrict__ Gi) {
  __shared__ float2 U[NW][2][2];
  const int tid = threadIdx.x;
  if (tid < NW) {
    float a  = params[tid]      * 0.5f;
    float bt = params[tid + NW] * 0.5f;
    float ca = cosf(a),  sa = sinf(a);
    float cb = cosf(bt), sb = sinf(bt);
    // U = RY(b) * RX(a)
    U[tid][0][0] = make_float2( cb * ca,  sb * sa);
    U[tid][0][1] = make_float2(-sb * ca, -cb * sa);
    U[tid][1][0] = make_float2( sb * ca, -cb * sa);
    U[tid][1][1] = make_float2( cb * ca, -sb * sa);
  }
  __syncthreads();
  const int trow = tid >> 4, scol = tid & 15;
  for (int p = 0; p < 5; ++p) {
    float gr = 1.0f, gi = 0.0f;
    #pragma unroll
    for (int d = 0; d < 4; ++d) {        // wire 4p+d <-> bit (3-d), MSB first
      int tb = (trow >> (3 - d)) & 1;
      int sb = (scol >> (3 - d)) & 1;
      float2 u = U[4 * p + d][tb][sb];
      float nr = gr * u.x - gi * u.y;
      float ni = gr * u.y + gi * u.x;
      gr = nr; gi = ni;
    }
    Gr[p * 256 + tid] = gr;
    Gi[p * 256 + tid] = gi;
  }
}

// ---------------------------------------------------------------------------
// Pass 1: wires 0..7 (bits 19..12).  WG = 16 consecutive lo x 256 q (32KB).
//   stage A: wires 4..7 (stride 2^12), real global -> LDS   (8 WMMA/tile)
//   stage B: wires 0..3 (stride 2^16), LDS -> complex global (16 WMMA/tile)
// ---------------------------------------------------------------------------
__global__ void pass1(const float* __restrict__ inR,
                      float2* __restrict__ outC,
                      const float* __restrict__ Gr,
                      const float* __restrict__ Gi) {
  __shared__ float2 lds[256 * 16];           // [q][j], j = lo - lo0
  const int lane = threadIdx.x & 31;
  const int wave = threadIdx.x >> 5;
  const int t    = lane & 15;
  const int hk   = (lane >> 4) << 1;         // 0 or 2
  const unsigned gbase = (blockIdx.y << 20) + (blockIdx.x << 4) + t;

  AFrag A;
  // ---- stage A ----
  loadA(Gr + 256, Gi + 256, t, hk, A);
  #pragma unroll
  for (int ii = 0; ii < 2; ++ii) {
    const int tn = wave + ii * 8;            // q_hi of this tile
    v2f Bre[4];
    #pragma unroll
    for (int k = 0; k < 4; ++k) {
      int r0 = 4 * k + hk;                   // q_lo rows
      float b0 = inR[gbase + ((unsigned)(tn * 16 + r0)     << 12)];
      float b1 = inR[gbase + ((unsigned)(tn * 16 + r0 + 1) << 12)];
      Bre[k].x = b0; Bre[k].y = b1;
    }
    v8f Dre = {}, Dim = {};
    #pragma unroll
    for (int k = 0; k < 4; ++k) Dre = wmma_f32(A.r[k], Bre[k], Dre);
    #pragma unroll
    for (int k = 0; k < 4; ++k) Dim = wmma_f32(A.i[k], Bre[k], Dim);
    #pragma unroll
    for (int r = 0; r < 8; ++r) {
      int row = r + (hk << 2);
      lds[(tn * 16 + row) * 16 + t] = make_float2(Dre[r], Dim[r]);
    }
  }
  __syncthreads();
  // ---- stage B ----
  loadA(Gr, Gi, t, hk, A);
  #pragma unroll
  for (int ii = 0; ii < 2; ++ii) {
    const int tn = wave + ii * 8;            // q_lo of this tile
    v2f Bre[4], Bim[4];
    #pragma unroll
    for (int k = 0; k < 4; ++k) {
      int r0 = 4 * k + hk;                   // q_hi rows
      float2 f0 = lds[(r0 * 16 + tn) * 16 + t];
      float2 f1 = lds[((r0 + 1) * 16 + tn) * 16 + t];
      Bre[k].x = f0.x; Bre[k].y = f1.x;
      Bim[k].x = f0.y; Bim[k].y = f1.y;
    }
    v8f Dre = {}, Dim = {};
    #pragma unroll
    for (int k = 0; k < 4; ++k) Dre = wmma_f32(A.r[k],  Bre[k], Dre);
    #pragma unroll
    for (int k = 0; k < 4; ++k) Dre = wmma_f32(A.ni[k], Bim[k], Dre);
    #pragma unroll
    for (int k = 0; k < 4; ++k) Dim = wmma_f32(A.i[k],  Bre[k], Dim);
    #pragma unroll
    for (int k = 0; k < 4; ++k) Dim = wmma_f32(A.r[k],  Bim[k], Dim);
    #pragma unroll
    for (int r = 0; r < 8; ++r) {
      int row = r + (hk << 2);
      outC[gbase + ((unsigned)(row * 16 + tn) << 12)] =
          make_float2(Dre[r], Dim[r]);
    }
  }
}

// ---------------------------------------------------------------------------
// Pass 2: wires 8..19 (bits 11..0).  WG = one contiguous 4096-el block (32KB).
//   stage 1: wires  8..11 (stride 256), global -> LDS
//   stage 2: wires 12..15 (stride  16), LDS in place
//   stage 3: wires 16..19 (stride   1), LDS -> fused signed |psi|^2 reduction
// ---------------------------------------------------------------------------
__global__ void pass2(const float2* __restrict__ inC,
                      const float* __restrict__ Gr,
                      const float* __restrict__ Gi,
                      float* __restrict__ partials) {
  __shared__ float2 lds[4096 + 256];         // skewed via SK()
  __shared__ float  red[256][NW];
  const int lane = threadIdx.x & 31;
  const int wave = threadIdx.x >> 5;
  const int t    = lane & 15;
  const int hk   = (lane >> 4) << 1;
  const unsigned blk  = blockIdx.x;          // [0,256) block within batch
  const unsigned base = (blockIdx.y << 20) + (blk << 12);

  AFrag A;
  // ---- stage 1: global -> LDS ----
  loadA(Gr + 512, Gi + 512, t, hk, A);
  #pragma unroll
  for (int ii = 0; ii < 2; ++ii) {
    const int tn = wave + ii * 8;
    v2f Bre[4], Bim[4];
    #pragma unroll
    for (int k = 0; k < 4; ++k) {
      int r0 = 4 * k + hk;                   // s = bits 11..8
      float2 f0 = inC[base + ((unsigned)r0       << 8) + tn * 16 + t];
      float2 f1 = inC[base + ((unsigned)(r0 + 1) << 8) + tn * 16 + t];
      Bre[k].x = f0.x; Bre[k].y = f1.x;
      Bim[k].x = f0.y; Bim[k].y = f1.y;
    }
    v8f Dre = {}, Dim = {};
    #pragma unroll
    for (int k = 0; k < 4; ++k) Dre = wmma_f32(A.r[k],  Bre[k], Dre);
    #pragma unroll
    for (int k = 0; k < 4; ++k) Dre = wmma_f32(A.ni[k], Bim[k], Dre);
    #pragma unroll
    for (int k = 0; k < 4; ++k) Dim = wmma_f32(A.i[k],  Bre[k], Dim);
    #pragma unroll
    for (int k = 0; k < 4; ++k) Dim = wmma_f32(A.r[k],  Bim[k], Dim);
    #pragma unroll
    for (int r = 0; r < 8; ++r) {
      int row = r + (hk << 2);
      lds[SK((row << 8) + tn * 16 + t)] = make_float2(Dre[r], Dim[r]);
    }
  }
  __syncthreads();
  // ---- stage 2: LDS in place ----
  loadA(Gr + 768, Gi + 768, t, hk, A);
  #pragma unroll
  for (int ii = 0; ii < 2; ++ii) {
    const int tn = wave + ii * 8;
    v2f Bre[4], Bim[4];
    #pragma unroll
    for (int k = 0; k < 4; ++k) {
      int r0 = 4 * k + hk;                   // s = bits 7..4
      unsigned idx = tn * 256 + r0 * 16 + t;
      float2 f0 = lds[SK(idx)];
      float2 f1 = lds[SK(idx + 16)];
      Bre[k].x = f0.x; Bre[k].y = f1.x;
      Bim[k].x = f0.y; Bim[k].y = f1.y;
    }
    v8f Dre = {}, Dim = {};
    #pragma unroll
    for (int k = 0; k < 4; ++k) Dre = wmma_f32(A.r[k],  Bre[k], Dre);
    #pragma unroll
    for (int k = 0; k < 4; ++k) Dre = wmma_f32(A.ni[k], Bim[k], Dre);
    #pragma unroll
    for (int k = 0; k < 4; ++k) Dim = wmma_f32(A.i[k],  Bre[k], Dim);
    #pragma unroll
    for (int k = 0; k < 4; ++k) Dim = wmma_f32(A.r[k],  Bim[k], Dim);
    #pragma unroll
    for (int r = 0; r < 8; ++r) {
      int row = r + (hk << 2);
      lds[SK(tn * 256 + row * 16 + t)] = make_float2(Dre[r], Dim[r]);
    }
  }
  __syncthreads();
  // ---- stage 3: LDS -> signed probability reduction ----
  loadA(Gr + 1024, Gi + 1024, t, hk, A);
  float acc[NW];
  #pragma unroll
  for (int w = 0; w < NW; ++w) acc[w] = 0.0f;
  #pragma unroll
  for (int ii = 0; ii < 2; ++ii) {
    const int tn = wave + ii * 8;
    const unsigned m = tn * 16 + t;          // bits 11..4 (passive)
    v2f Bre[4], Bim[4];
    #pragma unroll
    for (int k = 0; k < 4; ++k) {
      int r0 = 4 * k + hk;                   // s = bits 3..0
      float2 f0 = lds[SK(m * 16 + r0)];
      float2 f1 = lds[SK(m * 16 + r0 + 1)];
      Bre[k].x = f0.x; Bre[k].y = f1.x;
      Bim[k].x = f0.y; Bim[k].y = f1.y;
    }
    v8f Dre = {}, Dim = {};
    #pragma unroll
    for (int k = 0; k < 4; ++k) Dre = wmma_f32(A.r[k],  Bre[k], Dre);
    #pragma unroll
    for (int k = 0; k < 4; ++k) Dre = wmma_f32(A.ni[k], Bim[k], Dre);
    #pragma unroll
    for (int k = 0; k < 4; ++k) Dim = wmma_f32(A.i[k],  Bre[k], Dim);
    #pragma unroll
    for (int k = 0; k < 4; ++k) Dim = wmma_f32(A.r[k],  Bim[k], Dim);
    #pragma unroll
    for (int r = 0; r < 8; ++r) {
      int row = r + (hk << 2);
      unsigned idx = (blk << 12) + m * 16 + row;   // 20-bit basis index
      float p   = Dre[r] * Dre[r] + Dim[r] * Dim[r];
      float sgn = p;
      #pragma unroll
      for (int w = 0; w < NW; ++w) {         // wire w <-> bit (19-w)
        if (idx & (1u << (19 - w))) sgn = -sgn;
        acc[w] += sgn;
      }
    }
  }
  // deterministic fixed-order tree reduction (no float atomics)
  #pragma unroll
  for (int w = 0; w < NW; ++w) red[threadIdx.x][w] = acc[w];
  __syncthreads();
  for (int s = 128; s > 0; s >>= 1) {
    if ((int)threadIdx.x < s) {
      #pragma unroll
      for (int w = 0; w < NW; ++w)
        red[threadIdx.x][w] += red[threadIdx.x + s][w];
    }
    __syncthreads();
  }
  if (threadIdx.x < NW)
    partials[(blockIdx.y * gridDim.x + blk) * NW + threadIdx.x] =
        red[0][threadIdx.x];
}

// ---------------------------------------------------------------------------
__global__ void final_out(const float* __restrict__ partials,
                          const float* __restrict__ W,
                          const float* __restrict__ b0,
                          float* __restrict__ out, int wgs) {
  int b = threadIdx.x;
  if (b >= QBATCH) return;
  float s = b0[0];
  for (int i = 0; i < NW; ++i) {
    float f = 0.0f;
    for (int g = 0; g < wgs; ++g) f += partials[(b * wgs + g) * NW + i];
    s += W[i] * f;
  }
  out[b] = s;
}

// ---------------------------------------------------------------------------
extern "C" void kernel_launch(void* const* d_in, const int* in_sizes, int n_in,
                              void* d_out, int out_size, void* d_ws,
                              size_t ws_size, hipStream_t stream) {
  (void)in_sizes; (void)n_in; (void)out_size; (void)ws_size;
  const float* state  = (const float*)d_in[0];   // (32, 2^20) f32
  const float* params = (const float*)d_in[1];   // (40,)      f32
  const float* W      = (const float*)d_in[2];   // (1, 20)    f32
  const float* bsc    = (const float*)d_in[3];   // (1,)       f32

  // ws: [Gr 5*256][Gi 5*256][partials 32*256*20] ... @1MB: state (256MB)
  float*  Gr       = (float*)d_ws;
  float*  Gi       = Gr + 5 * 256;
  float*  partials = Gi + 5 * 256;
  float2* psi      = (float2*)((char*)d_ws + (1u << 20));

  setup_gates<<<1, 256, 0, stream>>>(params, Gr, Gi);

  dim3 grid(256, QBATCH);
  pass1<<<grid, 256, 0, stream>>>(state, psi, Gr, Gi);
  pass2<<<grid, 256, 0, stream>>>(psi, Gr, Gi, partials);

  final_out<<<1, 32, 0, stream>>>(partials, W, bsc, (float*)d_out, 256);
}